// FeedForward_36816459661927
// MI455X (gfx1250) — compile-verified
//
#include <hip/hip_runtime.h>
#include <math.h>
#include <stdint.h>

// ---------------- problem constants ----------------
#define NTOK 1024      // B*T
#define DIMX 512       // d_model
#define HIDX 2048      // d_ff
#define NEXP 64        // experts

typedef __attribute__((ext_vector_type(2))) float v2f;
typedef __attribute__((ext_vector_type(8))) float v8f;

// ---------------- CDNA5 async global->LDS copy (ASYNCcnt path) ----------------
__device__ __forceinline__ void async_g2l_b128(void* lds_ptr, const void* gptr) {
    uint32_t lds_off = (uint32_t)(uintptr_t)lds_ptr;          // LDS aperture: offset in addr[31:0]
    uint64_t gaddr   = (uint64_t)(uintptr_t)gptr;
    asm volatile("global_load_async_to_lds_b128 %0, %1, off"
                 :: "v"(lds_off), "v"(gaddr)
                 : "memory");
}
__device__ __forceinline__ void wait_async0() {
    asm volatile("s_wait_asynccnt 0" ::: "memory");
}

// ---------------- tiny helpers ----------------
__device__ __forceinline__ float gelu_erf(float x) {
    return 0.5f * x * (1.0f + erff(x * 0.70710678118654752440f));
}

// ---------------- kernel: zero expert counters ----------------
__global__ void zero_counts_kernel(int* cnt) {
    cnt[threadIdx.x] = 0;   // 128 ints: cnt1 (64) + cnt2 (64)
}

// ---------------- kernel: top-1 router ----------------
template <int DIN>
__global__ void router_kernel(const float* __restrict__ X,
                              const float* __restrict__ emb,
                              float* __restrict__ gate,
                              int* __restrict__ idx) {
    __shared__ float xb[DIN];
    __shared__ float lg[NEXP];
    const int n = blockIdx.x;
    const int t = threadIdx.x;          // 64 threads, one per expert
    for (int i = t; i < DIN; i += NEXP)
        xb[i] = X[(size_t)n * DIN + i];
    __syncthreads();

    const float* er = emb + (size_t)t * DIN;
    float s = 0.0f;
    #pragma unroll 4
    for (int i = 0; i < DIN; ++i) s += xb[i] * er[i];
    constexpr float scale = (DIN == 512) ? 0.044194173824159216f   // 1/sqrt(512)
                                         : 0.022097086912079608f;  // 1/sqrt(2048)
    lg[t] = s * scale;
    __syncthreads();

    if (t == 0) {
        float mx = lg[0]; int mi = 0;
        for (int i = 1; i < NEXP; ++i)
            if (lg[i] > mx) { mx = lg[i]; mi = i; }   // '>' keeps first index on ties
        float se = 0.0f;
        for (int i = 0; i < NEXP; ++i) se += expf(lg[i] - mx);
        gate[n] = 1.0f / se;
        idx[n]  = mi;
    }
}

// ---------------- kernel: bucket tokens by expert ----------------
__global__ void scatter_kernel(const int* __restrict__ idx,
                               int* __restrict__ cnt,
                               int* __restrict__ list) {
    int t = blockIdx.x * blockDim.x + threadIdx.x;
    if (t < NTOK) {
        int e = idx[t];
        int p = atomicAdd(&cnt[e], 1);
        list[e * NTOK + p] = t;
    }
}

// ---------------- kernel: grouped per-expert GEMM via fp32 WMMA ----------------
// Y[tok, col] = act( gate[tok] * ( X[tok,:] @ W[e,:,col] + b[e,col] ) )
// grid = (NEXP, DOUT/128); block = 256 threads = 8 wave32, each wave owns a 16-col tile.
// Double-buffered LDS, async global->LDS copies pipelined against WMMA compute.
template <int DIN, int DOUT, bool ACT_GELU>
__global__ __launch_bounds__(256)
void moe_gemm_kernel(const float* __restrict__ X,
                     const float* __restrict__ W,
                     const float* __restrict__ bias,
                     const float* __restrict__ gate,
                     const int* __restrict__ cnt,
                     const int* __restrict__ list,
                     float* __restrict__ Y) {
    constexpr int KC  = 64;          // K chunk staged in LDS
    constexpr int NT  = 128;         // output cols per workgroup (8 waves x 16)
    constexpr int XS  = 76;          // padded LDS row stride for X tile (16B aligned, bank-spread)
    constexpr int WS  = 136;         // padded LDS row stride for W tile
    constexpr int NKC = DIN / KC;    // # of K chunks (8 or 32, even)

    __shared__ float Xs[2][16 * XS];
    __shared__ float Ws[2][KC * WS];

    const int e    = blockIdx.x;
    const int nb   = blockIdx.y * NT;
    const int t    = threadIdx.x;
    const int lane = t & 31;
    const int wv   = t >> 5;        // wave id 0..7

    const int ce = cnt[e];
    if (ce == 0) return;            // wave-uniform: whole block exits together

    const float* We = W + (size_t)e * DIN * DOUT;
    const int* le   = list + e * NTOK;
    const int nchunks = (ce + 15) >> 4;

    // WMMA fragment coordinates (wave32, ISA 7.12.2 layouts)
    const int khalf = (lane >> 4) * 2;       // A/B: lanes 16-31 hold K+2,K+3
    const int arow  = lane & 15;             // A: M = lane
    const int wcol  = wv * 16 + (lane & 15); // B/D: N within this block's 128-col tile
    const int colg  = nb + wcol;             // global output column
    const float bv  = bias[(size_t)e * DOUT + colg];

    // thread -> staging coordinates (all lanes always active: no EXEC divergence)
    const int xr  = t >> 4;                  // X tile row 0..15
    const int xf4 = t & 15;                  // X tile float4 index 0..15

    v8f acc;

    // stage one K chunk (kc) into LDS buffer b via async copies
    auto stage = [&](int kc, int b, int m0) {
        // --- X tile: 16 rows x KC floats, gathered by token list (row-clamped) ---
        int mi = m0 + xr; if (mi >= ce) mi = ce - 1;   // clamp: duplicate rows never stored
        int tok = le[mi];
        async_g2l_b128(&Xs[b][xr * XS + xf4 * 4],
                       X + (size_t)tok * DIN + kc + xf4 * 4);
        // --- W tile: KC rows x 128 cols, fully coalesced ---
        #pragma unroll
        for (int i = 0; i < 8; ++i) {
            int ii  = i * 256 + t;
            int row = ii >> 5;
            int c4  = ii & 31;
            const float* gw = We + (size_t)(kc + row) * DOUT + nb + c4 * 4;
            async_g2l_b128(&Ws[b][row * WS + c4 * 4], gw);
            if (kc + 2 * KC < DIN)
                __builtin_prefetch(gw + (size_t)2 * KC * DOUT, 0, 1);  // chunk k+2 -> L2
        }
    };

    // consume one staged K chunk: register-batched fragments, then 16 chained WMMAs
    auto compute = [&](int b) {
        v2f af[KC / 4], bf[KC / 4];
        #pragma unroll
        for (int k4 = 0; k4 < KC / 4; ++k4) {
            int k = k4 * 4;
            af[k4].x = Xs[b][arow * XS + k + khalf];
            af[k4].y = Xs[b][arow * XS + k + khalf + 1];
            bf[k4].x = Ws[b][(k + khalf) * WS + wcol];
            bf[k4].y = Ws[b][(k + khalf + 1) * WS + wcol];
        }
        #pragma unroll
        for (int k4 = 0; k4 < KC / 4; ++k4) {
            acc = __builtin_amdgcn_wmma_f32_16x16x4_f32(
                    /*neg_a=*/false, af[k4], /*neg_b=*/false, bf[k4],
                    /*c_mod=*/(short)0, acc,
                    /*reuse_a=*/false, /*reuse_b=*/false);
        }
    };

    for (int mc = 0; mc < nchunks; ++mc) {
        const int m0 = mc * 16;
        acc = (v8f){};

        // prologue: fill buffer 0
        stage(0, 0, m0);
        wait_async0();
        __syncthreads();

        // pipelined main loop: async-fill next buffer while computing current
        for (int ki = 0; ki < NKC; ++ki) {
            const int cur = ki & 1;
            if (ki + 1 < NKC) stage((ki + 1) * KC, cur ^ 1, m0);
            compute(cur);
            wait_async0();      // next buffer resident (per-wave) ...
            __syncthreads();    // ... and across all waves
        }

        // --- epilogue: bias, gate, activation, scatter-store ---
        const int mrow0 = (lane >> 4) * 8;   // D: lanes 16-31 hold M=8..15
        #pragma unroll
        for (int v = 0; v < 8; ++v) {
            int mi = m0 + mrow0 + v;
            if (mi < ce) {
                int tok = le[mi];
                float val = (acc[v] + bv) * gate[tok];
                float y = ACT_GELU ? gelu_erf(val) : val;
                Y[(size_t)tok * DOUT + colg] = y;
            }
        }
    }
}

// ---------------- host launcher ----------------
extern "C" void kernel_launch(void* const* d_in, const int* in_sizes, int n_in,
                              void* d_out, int out_size, void* d_ws, size_t ws_size,
                              hipStream_t stream) {
    const float* x    = (const float*)d_in[0];   // [1024, 512]
    const float* emb1 = (const float*)d_in[1];   // [64, 512]
    const float* W1   = (const float*)d_in[2];   // [64, 512, 2048]
    const float* b1   = (const float*)d_in[3];   // [64, 2048]
    const float* emb2 = (const float*)d_in[4];   // [64, 2048]
    const float* W2   = (const float*)d_in[5];   // [64, 2048, 512]
    const float* b2   = (const float*)d_in[6];   // [64, 512]
    float* out = (float*)d_out;                  // [1024, 512]

    // workspace carve-up
    char* ws = (char*)d_ws;
    float* h     = (float*)ws;                   ws += (size_t)NTOK * HIDX * sizeof(float);
    float* gate1 = (float*)ws;                   ws += NTOK * sizeof(float);
    int*   idx1  = (int*)ws;                     ws += NTOK * sizeof(int);
    float* gate2 = (float*)ws;                   ws += NTOK * sizeof(float);
    int*   idx2  = (int*)ws;                     ws += NTOK * sizeof(int);
    int*   cnt   = (int*)ws;                     ws += 128 * sizeof(int);   // cnt1[64] + cnt2[64]
    int*   list1 = (int*)ws;                     ws += (size_t)NEXP * NTOK * sizeof(int);
    int*   list2 = (int*)ws;
    int* cnt1 = cnt;
    int* cnt2 = cnt + NEXP;

    zero_counts_kernel<<<1, 128, 0, stream>>>(cnt);

    // ---- layer 1 ----
    router_kernel<DIMX><<<NTOK, NEXP, 0, stream>>>(x, emb1, gate1, idx1);
    scatter_kernel<<<(NTOK + 255) / 256, 256, 0, stream>>>(idx1, cnt1, list1);
    moe_gemm_kernel<DIMX, HIDX, true>
        <<<dim3(NEXP, HIDX / 128), 256, 0, stream>>>(x, W1, b1, gate1, cnt1, list1, h);

    // ---- layer 2 ----
    router_kernel<HIDX><<<NTOK, NEXP, 0, stream>>>(h, emb2, gate2, idx2);
    scatter_kernel<<<(NTOK + 255) / 256, 256, 0, stream>>>(idx2, cnt2, list2);
    moe_gemm_kernel<HIDX, DIMX, false>
        <<<dim3(NEXP, DIMX / 128), 256, 0, stream>>>(h, W2, b2, gate2, cnt2, list2, out);
}